// StochasticDownsampler_9620726743160
// MI455X (gfx1250) — compile-verified
//
#include <hip/hip_runtime.h>
#include <hip/hip_bf16.h>

// Problem constants (from reference setup_inputs)
#define B_   4
#define C_   16
#define H_   512
#define W_   512
#define SPP_ 16
#define RH_  256
#define RW_  256

// Tiling
#define TILE   16            // 16x16 output pixels per workgroup
#define CH     8             // channels per workgroup (2 groups cover C=16)
#define ROWS   34            // input rows staged:   [2*rh0-1 .. 2*rh0+32]
#define LDSW   36            // input cols staged:   [2*rw0-2 .. 2*rw0+33]
#define NTHREADS 256

__device__ __forceinline__ void fma4(float* acc, float w, const float4 v) {
    acc[0] += w * v.x; acc[1] += w * v.y; acc[2] += w * v.z; acc[3] += w * v.w;
}

__global__ __launch_bounds__(NTHREADS)
void stochastic_downsample_kernel(const float* __restrict__ x,
                                  const float* __restrict__ jitter,
                                  float* __restrict__ out) {
    // LDS tile: [ROWS][LDSW][CH] floats, channel-interleaved so one (y,x) texel
    // is 8 contiguous floats (32B) -> two ds_load_b128 per bilinear tap.
    __shared__ __align__(16) float tile[ROWS * LDSW * CH];   // 39,168 B

    const int tid  = threadIdx.x;
    const int bz   = blockIdx.z;
    const int b    = bz >> 1;
    const int c0   = (bz & 1) * CH;
    const int rw0  = blockIdx.x * TILE;
    const int rh0  = blockIdx.y * TILE;
    const int ox   = 2 * rw0 - 2;   // global col of LDS col 0
    const int oy   = 2 * rh0 - 1;   // global row of LDS row 0

    const float* xb = x + (size_t)(b * C_ + c0) * (size_t)(H_ * W_);

    // ---- Stage input footprint into LDS with async global->LDS copies ----
    // Per-element border clamp on the SOURCE address exactly reproduces the
    // reference's coordinate clipping (clamped taps also carry weight 0).
    const int TOTAL = ROWS * LDSW * CH;   // 9792 lane-copies
    for (int id = tid; id < TOTAL; id += NTHREADS) {
        const int xcol = id % LDSW;
        const int rem  = id / LDSW;
        const int r    = rem % ROWS;
        const int c    = rem / ROWS;
        const int gy   = min(max(oy + r, 0), H_ - 1);
        const int gx   = min(max(ox + xcol, 0), W_ - 1);
        const float* gp = xb + (size_t)c * (size_t)(H_ * W_)
                             + (size_t)gy * W_ + gx;
        const unsigned loff =
            (unsigned)(size_t)&tile[(r * LDSW + xcol) * CH + c];
        asm volatile("global_load_async_to_lds_b32 %0, %1, off"
                     :: "v"(loff), "v"((unsigned long long)(size_t)gp)
                     : "memory");
    }

    // ---- Prefetch all jitter samples into registers WHILE the async LDS
    // fill is in flight (LOADcnt vs ASYNCcnt are independent counters). ----
    const int px = tid & (TILE - 1);
    const int py = tid >> 4;
    const int rw = rw0 + px;
    const int rh = rh0 + py;

    const float* jbase = jitter + ((size_t)rh * RW_ + rw) * 2;
    float2 jit[SPP_];
#pragma unroll
    for (int s = 0; s < SPP_; ++s) {
        jit[s] = *(const float2*)(jbase + (size_t)s * (RH_ * RW_ * 2));
    }

    // Wait for this wave's async copies, then sync the workgroup.
    asm volatile("s_wait_asynccnt 0" ::: "memory");
    __syncthreads();

    // ---- Sample: one thread per output pixel, 8 channels in registers ----
    float acc[CH];
#pragma unroll
    for (int c = 0; c < CH; ++c) acc[c] = 0.0f;

#pragma unroll
    for (int s = 0; s < SPP_; ++s) {
        const float2 j = jit[s];
        // ix = clip(((gx+1)*W - 1)/2) == clip(2*rw + 2*jx - 0.5, 0, W-1)
        float ix = fminf(fmaxf(2.0f * (float)rw + 2.0f * j.x - 0.5f, 0.0f),
                         (float)(W_ - 1));
        float iy = fminf(fmaxf(2.0f * (float)rh + 2.0f * j.y - 0.5f, 0.0f),
                         (float)(H_ - 1));
        const float fx = floorf(ix);
        const float fy = floorf(iy);
        const float wx1 = ix - fx;
        const float wy1 = iy - fy;
        // Clamped LDS indices; the +1 taps past a clamp have weight exactly 0.
        const int lx0 = min(max((int)fx - ox, 0), LDSW - 2);
        const int ly0 = min(max((int)fy - oy, 0), ROWS - 2);
        const float w00 = (1.0f - wy1) * (1.0f - wx1);
        const float w01 = (1.0f - wy1) * wx1;
        const float w10 = wy1 * (1.0f - wx1);
        const float w11 = wy1 * wx1;

        const float* t00 = &tile[(ly0 * LDSW + lx0) * CH];
        const float4* p00 = (const float4*)t00;
        const float4* p01 = p00 + (CH / 4);                    // lx0+1
        const float4* p10 = (const float4*)(t00 + LDSW * CH);  // ly0+1
        const float4* p11 = p10 + (CH / 4);

        fma4(acc,     w00, p00[0]); fma4(acc + 4, w00, p00[1]);
        fma4(acc,     w01, p01[0]); fma4(acc + 4, w01, p01[1]);
        fma4(acc,     w10, p10[0]); fma4(acc + 4, w10, p10[1]);
        fma4(acc,     w11, p11[0]); fma4(acc + 4, w11, p11[1]);
    }

    // ---- Write out: mean over spp ----
    const float scale = 1.0f / (float)SPP_;
#pragma unroll
    for (int c = 0; c < CH; ++c) {
        out[((size_t)(b * C_ + c0 + c) * RH_ + rh) * RW_ + rw] = acc[c] * scale;
    }
}

extern "C" void kernel_launch(void* const* d_in, const int* in_sizes, int n_in,
                              void* d_out, int out_size, void* d_ws, size_t ws_size,
                              hipStream_t stream) {
    (void)in_sizes; (void)n_in; (void)out_size; (void)d_ws; (void)ws_size;
    const float* x      = (const float*)d_in[0];
    const float* jitter = (const float*)d_in[1];
    float* out          = (float*)d_out;

    dim3 grid(RW_ / TILE, RH_ / TILE, B_ * (C_ / CH));   // 16 x 16 x 8
    dim3 block(NTHREADS);
    stochastic_downsample_kernel<<<grid, block, 0, stream>>>(x, jitter, out);
}